// DifferentiableRiskBudgeting_54984171323874
// MI455X (gfx1250) — compile-verified
//
#include <hip/hip_runtime.h>

// ---------------------------------------------------------------------------
// Differentiable risk budgeting, B=32768 batches, P=45.
// One wave32 per batch. Sigma (45x45, padded 48x48) held in registers as
// V_WMMA_F32_16X16X4_F32 A-fragments (36 x float2 per lane), loaded from HBM
// exactly once. 20 projected-gradient iterations run fully on-chip:
// matvec on the WMMA pipe (f32, full precision), projection on VALU with
// wave32 shuffle reductions.
// ---------------------------------------------------------------------------

typedef __attribute__((ext_vector_type(2))) float v2f;
typedef __attribute__((ext_vector_type(8))) float v8f;

#define P_DIM    45
#define P_PAD    48
#define N_ITER   20
#define STEP     0.05f
#define MAXW     0.15f
#define EPSILON  1e-8f
#define WPB      8   // waves (batches) per block

__device__ __forceinline__ float sgnf(float x) {
    return (x > 0.0f) ? 1.0f : ((x < 0.0f) ? -1.0f : 0.0f);
}

__device__ __forceinline__ float clip_w(float x) {
    return fminf(fmaxf(x, 0.0f), MAXW);
}

__device__ __forceinline__ float wave_sum(float v) {
    #pragma unroll
    for (int off = 16; off > 0; off >>= 1)
        v += __shfl_xor(v, off, 32);
    return v;
}

__global__ __launch_bounds__(256) void
risk_budget_wmma_kernel(const float* __restrict__ sigma,
                        const float* __restrict__ beta,
                        const float* __restrict__ w_prev,
                        const float* __restrict__ log_ls,
                        const float* __restrict__ log_lt,
                        float* __restrict__ out,
                        int nB)
{
    __shared__ float wsh[WPB][P_PAD];   // current w, padded with zeros
    __shared__ float swsh[WPB][P_PAD];  // Sigma @ w scratch

    const int lane = threadIdx.x & 31;
    const int slot = threadIdx.x >> 5;
    const int b    = blockIdx.x * WPB + slot;
    if (b >= nB) return;   // uniform per wave -> EXEC all-ones inside

    const float lam_s = expf(log_ls[0]);
    const float lam_t = expf(log_lt[0]);

    // Row ownership: lane owns rows rA = lane, rB = lane + 32 (if < 45).
    const int  rA   = lane;
    const int  rB   = lane + 32;
    const bool hasB = (rB < P_DIM);

    const float betaA  = beta  [(size_t)b * P_DIM + rA];
    const float wprevA = w_prev[(size_t)b * P_DIM + rA];
    const float betaB  = hasB ? beta  [(size_t)b * P_DIM + rB] : 0.0f;
    const float wprevB = hasB ? w_prev[(size_t)b * P_DIM + rB] : 0.0f;

    // ---- Load Sigma once from HBM into WMMA A-fragments (registers). ----
    // f32 16x16x4 A layout: lane (h = lane>>4, ml = lane&15) holds
    // A[ml, 2h+0] in VGPR0(.x) and A[ml, 2h+1] in VGPR1(.y).
    const float* sig = sigma + (size_t)b * (P_DIM * P_DIM);
    const int ml = lane & 15;
    const int h2 = (lane >> 4) << 1;
    v2f At[36];
    #pragma unroll
    for (int blk = 0; blk < 3; ++blk) {
        #pragma unroll
        for (int kc = 0; kc < 12; ++kc) {
            const int m  = blk * 16 + ml;
            const int k0 = kc * 4 + h2;
            v2f a;
            a.x = (m < P_DIM && k0     < P_DIM) ? sig[m * P_DIM + k0    ] : 0.0f;
            a.y = (m < P_DIM && k0 + 1 < P_DIM) ? sig[m * P_DIM + k0 + 1] : 0.0f;
            At[blk * 12 + kc] = a;
        }
    }

    // ---- Initialize w = 1/P, zero the pad lanes of wsh. ----
    float wA = 1.0f / (float)P_DIM;
    float wB = 1.0f / (float)P_DIM;
    wsh[slot][rA] = wA;
    if (hasB) wsh[slot][rB] = wB;
    if (lane >= 13 && lane < 16) wsh[slot][lane + 32] = 0.0f;  // rows 45..47
    __builtin_amdgcn_wave_barrier();

    // ---- 20 projected-gradient iterations, fully on-chip. ----
    #pragma unroll 1
    for (int it = 0; it < N_ITER; ++it) {
        // Sw = Sigma @ w via 3 row-blocks x 12 K-chunks of f32 WMMA.
        // B operand: w[k] broadcast into all 16 columns.
        #pragma unroll
        for (int blk = 0; blk < 3; ++blk) {
            v8f acc = {0.f, 0.f, 0.f, 0.f, 0.f, 0.f, 0.f, 0.f};
            #pragma unroll
            for (int kc = 0; kc < 12; ++kc) {
                const int kb = kc * 4 + h2;
                v2f bv;
                bv.x = wsh[slot][kb];
                bv.y = wsh[slot][kb + 1];
                acc = __builtin_amdgcn_wmma_f32_16x16x4_f32(
                    /*neg_a=*/false, At[blk * 12 + kc],
                    /*neg_b=*/false, bv,
                    /*c_mod=*/(short)0, acc,
                    /*reuse_a=*/false, /*reuse_b=*/false);
            }
            // D columns are replicated: lane 0 holds rows blk*16+0..7 in its
            // 8 accumulators, lane 16 holds rows blk*16+8..15. Bounce to LDS.
            if ((lane & 15) == 0) {
                const int base = blk * 16 + (lane >> 4) * 8;
                #pragma unroll
                for (int j = 0; j < 8; ++j) swsh[slot][base + j] = acc[j];
            }
        }
        __builtin_amdgcn_wave_barrier();

        const float SwA = swsh[slot][rA];
        const float SwB = hasB ? swsh[slot][rB] : 0.0f;

        // Gradient step.
        float nwA = wA - STEP * (2.0f * SwA - betaA + lam_s * sgnf(wA)
                                 + 2.0f * lam_t * (wA - wprevA));
        float nwB = wB - STEP * (2.0f * SwB - betaB + lam_s * sgnf(wB)
                                 + 2.0f * lam_t * (wB - wprevB));

        // Simplex projection: (clip, renorm) x 2.
        nwA = clip_w(nwA);
        nwB = clip_w(nwB);
        float s = wave_sum(nwA + (hasB ? nwB : 0.0f));
        float d = s + EPSILON;
        nwA = nwA / d;
        nwB = nwB / d;

        nwA = clip_w(nwA);
        nwB = clip_w(nwB);
        s = wave_sum(nwA + (hasB ? nwB : 0.0f));
        d = s + EPSILON;
        wA = nwA / d;
        wB = nwB / d;

        // Publish w for next iteration's B-operand broadcasts.
        wsh[slot][rA] = wA;
        if (hasB) wsh[slot][rB] = wB;
        __builtin_amdgcn_wave_barrier();
    }

    // ---- Write result. ----
    out[(size_t)b * P_DIM + rA] = wA;
    if (hasB) out[(size_t)b * P_DIM + rB] = wB;
}

extern "C" void kernel_launch(void* const* d_in, const int* in_sizes, int n_in,
                              void* d_out, int out_size, void* d_ws, size_t ws_size,
                              hipStream_t stream) {
    const float* sigma  = (const float*)d_in[0];
    const float* beta   = (const float*)d_in[1];
    const float* w_prev = (const float*)d_in[2];
    const float* lls    = (const float*)d_in[3];
    const float* llt    = (const float*)d_in[4];
    float* out = (float*)d_out;

    const int nB   = in_sizes[1] / P_DIM;      // beta is [B, 45]
    const int grid = (nB + WPB - 1) / WPB;

    hipLaunchKernelGGL(risk_budget_wmma_kernel, dim3(grid), dim3(256), 0, stream,
                       sigma, beta, w_prev, lls, llt, out, nB);
}